// DistanceAwareSelfAttentionHead_31190052503639
// MI455X (gfx1250) — compile-verified
//
#include <hip/hip_runtime.h>
#include <math.h>

// ---------------------------------------------------------------------------
// DistanceAwareSelfAttentionHead for MI455X (gfx1250, wave32, WMMA).
//
// Pipeline (all fp32, native v_wmma_f32_16x16x4_f32 tensor ops):
//   1. K = x@Wk, Q = x@Wq, V = x@Wv            (WMMA GEMM, NN)
//   2. A = 2 * Q @ K^T                          (WMMA GEMM, NT)
//   3. per-node dots qEk0/qEk1/kEq0/kEq1        (rank-2 edge-bias trick)
//   4. per-edge bias scatter-add into A         (global_atomic_add_f32)
//   5. row softmax of A/sqrt(512)               (64MB A lives in 192MB L2)
//   6. c0[i]=sum_e A*s, c1[i]=sum_e A*(1-s)     (atomic per-edge gather)
//   7. M = A @ V                                (WMMA GEMM, NN)
//   8. M += c0 x Ev[0] + c1 x Ev[1]             (rank-2 epilogue)
// ---------------------------------------------------------------------------

typedef __attribute__((ext_vector_type(2))) float v2f;
typedef __attribute__((ext_vector_type(8))) float v8f;

#define N_NODES 4096
#define FEAT    512
#define HID     256
#define N_EDGES 131072

// ---------------------------------------------------------------------------
// Tiled fp32 WMMA GEMM.  C[M,Nc] = alpha * A[M,Kd] @ op(B), all row-major.
//   BT=1: op(B) = B^T with B stored [Nc,Kd]  (NT: used for Q @ K^T)
//   BT=0: op(B) = B   with B stored [Kd,Nc]  (NN: projections, A @ V)
// Block = 256 threads = 8 wave32s, arranged 4(M) x 2(N); each wave owns a
// 16x64 strip (4 x v8f accumulators), block tile = 64(M) x 128(N).
// V_WMMA_F32_16X16X4_F32 fragment layout (ISA 7.12.2):
//   A 16x4 : lanes 0-15 M=0..15 hold K={0,1}; lanes 16-31 hold K={2,3}
//   B 4x16 : lanes 0-15 N=0..15 hold K={0,1}; lanes 16-31 hold K={2,3}
//   C 16x16: VGPR r -> M=r (lanes 0-15, N=lane) / M=r+8 (lanes 16-31)
// All tile dims divide the problem sizes exactly, so EXEC stays all-ones
// through every WMMA (wave-uniform guard only).
// ---------------------------------------------------------------------------
template <int BT>
__global__ __launch_bounds__(256) void wmma_gemm_f32(
    const float* __restrict__ A, const float* __restrict__ B,
    float* __restrict__ C, int M, int Nc, int Kd, float alpha)
{
  const int lane    = threadIdx.x & 31;
  const int wave    = threadIdx.x >> 5;
  const int wm      = wave & 3;                 // 0..3 (M direction)
  const int wn      = wave >> 2;                // 0..1 (N direction)
  const int rowBase = blockIdx.y * 64 + wm * 16;
  const int colBase = blockIdx.x * 128 + wn * 64;
  if (rowBase >= M || colBase >= Nc) return;    // wave-uniform

  const int half = lane >> 4;                   // 0: K={0,1}, 1: K={2,3}
  const int l15  = lane & 15;

  v8f acc[4] = {};

  const float* aRow = A + (size_t)(rowBase + l15) * Kd + half * 2;

  for (int k = 0; k < Kd; k += 4) {
    if (k + 64 < Kd) __builtin_prefetch(aRow + k + 64, 0, 0);
    const v2f afrag = *(const v2f*)(aRow + k);
#pragma unroll
    for (int nt = 0; nt < 4; ++nt) {
      const int col = colBase + nt * 16 + l15;
      v2f bfrag;
      if (BT) {
        bfrag = *(const v2f*)(B + (size_t)col * Kd + k + half * 2);
      } else {
        const int kk = k + half * 2;
        bfrag.x = B[(size_t)kk * Nc + col];
        bfrag.y = B[(size_t)(kk + 1) * Nc + col];
      }
      acc[nt] = __builtin_amdgcn_wmma_f32_16x16x4_f32(
          /*neg_a=*/false, afrag, /*neg_b=*/false, bfrag,
          /*c_mod=*/(short)0, acc[nt], /*reuse_a=*/false, /*reuse_b=*/false);
    }
  }

#pragma unroll
  for (int nt = 0; nt < 4; ++nt) {
    const int col = colBase + nt * 16 + l15;
#pragma unroll
    for (int r = 0; r < 8; ++r) {
      const int m = rowBase + r + half * 8;
      C[(size_t)m * Nc + col] = alpha * acc[nt][r];
    }
  }
}

// ---------------------------------------------------------------------------
// Per-node embedding dot products: one block (256 threads) per node.
//   qEk0[i]=q_i.Ek[0]  qEk1[i]=q_i.Ek[1]  kEq0[i]=k_i.Eq[0]  kEq1[i]=k_i.Eq[1]
// ---------------------------------------------------------------------------
__global__ __launch_bounds__(256) void node_dots_kernel(
    const float* __restrict__ q, const float* __restrict__ k,
    const float* __restrict__ Ek, const float* __restrict__ Eq,
    float* __restrict__ qEk0, float* __restrict__ qEk1,
    float* __restrict__ kEq0, float* __restrict__ kEq1)
{
  __shared__ float red[4][256];
  const int i = blockIdx.x;
  const int t = threadIdx.x;
  const float qv = q[(size_t)i * HID + t];
  const float kv = k[(size_t)i * HID + t];
  red[0][t] = qv * Ek[t];
  red[1][t] = qv * Ek[HID + t];
  red[2][t] = kv * Eq[t];
  red[3][t] = kv * Eq[HID + t];
  __syncthreads();
  for (int s = 128; s > 0; s >>= 1) {
    if (t < s) {
#pragma unroll
      for (int j = 0; j < 4; ++j) red[j][t] += red[j][t + s];
    }
    __syncthreads();
  }
  if (t == 0) {
    qEk0[i] = red[0][0]; qEk1[i] = red[1][0];
    kEq0[i] = red[2][0]; kEq1[i] = red[3][0];
  }
}

// ---------------------------------------------------------------------------
// Per-edge bias: s = sigmoid((attr - bias)*mult);
// A[src,dst] += s*qEk0[src] + (1-s)*qEk1[src] + s*kEq0[dst] + (1-s)*kEq1[dst]
// (atomic: duplicate edges must sum, matching A.at[src,dst].add)
// ---------------------------------------------------------------------------
__global__ __launch_bounds__(256) void edge_bias_kernel(
    const int* __restrict__ ei, const float* __restrict__ attr,
    const float* __restrict__ bias_p, const float* __restrict__ mult_p,
    const float* __restrict__ qEk0, const float* __restrict__ qEk1,
    const float* __restrict__ kEq0, const float* __restrict__ kEq1,
    float* __restrict__ A, float* __restrict__ s_out)
{
  const int e = blockIdx.x * blockDim.x + threadIdx.x;
  if (e >= N_EDGES) return;
  const int src = ei[e];
  const int dst = ei[N_EDGES + e];
  const float s = 1.0f / (1.0f + __expf(-(attr[e] - bias_p[0]) * mult_p[0]));
  s_out[e] = s;
  const float eb = s * qEk0[src] + (1.0f - s) * qEk1[src] +
                   s * kEq0[dst] + (1.0f - s) * kEq1[dst];
  atomicAdd(&A[(size_t)src * N_NODES + dst], eb);
}

// ---------------------------------------------------------------------------
// Row softmax of A/sqrt(FEAT): one block (256 threads) per row, 16 elems/thread.
// ---------------------------------------------------------------------------
__global__ __launch_bounds__(256) void softmax_row_kernel(float* __restrict__ A)
{
  __shared__ float red[256];
  const int row = blockIdx.x;
  const int t   = threadIdx.x;
  float* rp = A + (size_t)row * N_NODES;
  const float scale = 0.04419417382415922f;   // 1/sqrt(512)

  float vals[16];
  float m = -3.402823466e+38f;
#pragma unroll
  for (int j = 0; j < 16; ++j) {
    const float v = rp[t + j * 256] * scale;
    vals[j] = v;
    m = fmaxf(m, v);
  }
  red[t] = m; __syncthreads();
  for (int s = 128; s > 0; s >>= 1) {
    if (t < s) red[t] = fmaxf(red[t], red[t + s]);
    __syncthreads();
  }
  m = red[0]; __syncthreads();

  float sum = 0.0f;
#pragma unroll
  for (int j = 0; j < 16; ++j) { vals[j] = __expf(vals[j] - m); sum += vals[j]; }
  red[t] = sum; __syncthreads();
  for (int s = 128; s > 0; s >>= 1) {
    if (t < s) red[t] += red[t + s];
    __syncthreads();
  }
  const float inv = 1.0f / red[0];
#pragma unroll
  for (int j = 0; j < 16; ++j) rp[t + j * 256] = vals[j] * inv;
}

// ---------------------------------------------------------------------------
// c0[src] += A[src,dst]*s ; c1[src] += A[src,dst]*(1-s)  (post-softmax A)
// ---------------------------------------------------------------------------
__global__ __launch_bounds__(256) void edge_gather_kernel(
    const int* __restrict__ ei, const float* __restrict__ s_e,
    const float* __restrict__ A, float* __restrict__ c0, float* __restrict__ c1)
{
  const int e = blockIdx.x * blockDim.x + threadIdx.x;
  if (e >= N_EDGES) return;
  const int src = ei[e];
  const int dst = ei[N_EDGES + e];
  const float a = A[(size_t)src * N_NODES + dst];
  const float s = s_e[e];
  atomicAdd(&c0[src], a * s);
  atomicAdd(&c1[src], a * (1.0f - s));
}

__global__ __launch_bounds__(256) void zero_kernel(float* __restrict__ p, int n)
{
  const int i = blockIdx.x * blockDim.x + threadIdx.x;
  if (i < n) p[i] = 0.0f;
}

// M[i,j] += c0[i]*Ev[0,j] + c1[i]*Ev[1,j]
__global__ __launch_bounds__(256) void rank2_add_kernel(
    float* __restrict__ Mout, const float* __restrict__ c0,
    const float* __restrict__ c1, const float* __restrict__ Ev)
{
  const int idx = blockIdx.x * blockDim.x + threadIdx.x;  // over N*FEAT
  const int i = idx >> 9;         // FEAT = 512
  const int j = idx & 511;
  Mout[idx] += c0[i] * Ev[j] + c1[i] * Ev[FEAT + j];
}

// ---------------------------------------------------------------------------
extern "C" void kernel_launch(void* const* d_in, const int* in_sizes, int n_in,
                              void* d_out, int out_size, void* d_ws, size_t ws_size,
                              hipStream_t stream)
{
  const float* x    = (const float*)d_in[0];   // [N, FEAT]
  const int*   ei   = (const int*)  d_in[1];   // [2, E]
  const float* attr = (const float*)d_in[2];   // [E, 1]
  const float* Wk   = (const float*)d_in[3];   // [FEAT, HID]
  const float* Wq   = (const float*)d_in[4];   // [FEAT, HID]
  const float* Wv   = (const float*)d_in[5];   // [FEAT, FEAT]
  const float* Ek   = (const float*)d_in[6];   // [2, HID]
  const float* Eq   = (const float*)d_in[7];   // [2, HID]
  const float* Ev   = (const float*)d_in[8];   // [2, FEAT]
  const float* bias = (const float*)d_in[9];   // [1]
  const float* mult = (const float*)d_in[10];  // [1]
  float* Mout = (float*)d_out;                 // [N, FEAT]

  // Workspace layout (~81 MB; A alone is 64 MB and fits in the 192 MB L2).
  char* ws = (char*)d_ws;
  float* Q    = (float*)(ws);                                  // 4 MB
  float* K    = (float*)(ws + ((size_t)4  << 20));             // 4 MB
  float* V    = (float*)(ws + ((size_t)8  << 20));             // 8 MB
  float* A    = (float*)(ws + ((size_t)16 << 20));             // 64 MB
  float* qEk0 = (float*)(ws + ((size_t)80 << 20));
  float* qEk1 = qEk0 + N_NODES;
  float* kEq0 = qEk1 + N_NODES;
  float* kEq1 = kEq0 + N_NODES;
  float* c0   = kEq1 + N_NODES;
  float* c1   = c0   + N_NODES;
  float* s_e  = c1   + N_NODES;                                // E floats

  const dim3 blk(256);

  // 1. projections (NN GEMMs)
  wmma_gemm_f32<0><<<dim3(HID / 128,  N_NODES / 64), blk, 0, stream>>>(
      x, Wk, K, N_NODES, HID, FEAT, 1.0f);
  wmma_gemm_f32<0><<<dim3(HID / 128,  N_NODES / 64), blk, 0, stream>>>(
      x, Wq, Q, N_NODES, HID, FEAT, 1.0f);
  wmma_gemm_f32<0><<<dim3(FEAT / 128, N_NODES / 64), blk, 0, stream>>>(
      x, Wv, V, N_NODES, FEAT, FEAT, 1.0f);

  // 2. A = 2 * Q @ K^T (NT GEMM)
  wmma_gemm_f32<1><<<dim3(N_NODES / 128, N_NODES / 64), blk, 0, stream>>>(
      Q, K, A, N_NODES, N_NODES, HID, 2.0f);

  // 3. per-node embedding dots
  node_dots_kernel<<<N_NODES, 256, 0, stream>>>(Q, K, Ek, Eq,
                                                qEk0, qEk1, kEq0, kEq1);

  // 4. per-edge bias scatter into A
  edge_bias_kernel<<<N_EDGES / 256, 256, 0, stream>>>(
      ei, attr, bias, mult, qEk0, qEk1, kEq0, kEq1, A, s_e);

  // 5. row softmax
  softmax_row_kernel<<<N_NODES, 256, 0, stream>>>(A);

  // 6. rank-2 edge coefficients
  zero_kernel<<<(2 * N_NODES) / 256, 256, 0, stream>>>(c0, 2 * N_NODES);
  edge_gather_kernel<<<N_EDGES / 256, 256, 0, stream>>>(ei, s_e, A, c0, c1);

  // 7. M = A @ V (NN GEMM)
  wmma_gemm_f32<0><<<dim3(FEAT / 128, N_NODES / 64), blk, 0, stream>>>(
      A, V, Mout, N_NODES, FEAT, N_NODES, 1.0f);

  // 8. rank-2 epilogue
  rank2_add_kernel<<<((size_t)N_NODES * FEAT) / 256, 256, 0, stream>>>(
      Mout, c0, c1, Ev);
}